// SimpleRasterizer_26113401160431
// MI455X (gfx1250) — compile-verified
//
#include <hip/hip_runtime.h>
#include <stdint.h>

// ---- CDNA5 vector types ----
typedef float v2f __attribute__((ext_vector_type(2)));
typedef float v8f __attribute__((ext_vector_type(8)));
typedef unsigned int v4u __attribute__((ext_vector_type(4)));
typedef int v4i __attribute__((ext_vector_type(4)));
typedef int v8i __attribute__((ext_vector_type(8)));

#define HH 128
#define WW 128
#define KK 8
#define EPSF 1e-8f
#define BIGF 1e10f
#define FPAD_MAX 1008   // 63 tiles of 16 faces

// ---------------------------------------------------------------------------
// Kernel 1: per-face preprocessing.
// ws layout (float4, stride FPAD): [edge0 coefs][edge1 coefs][edge2 coefs][z0,z1,z2,invArea]
// edge coef float4 = (Cx, Cy, C0, 0) so that w = Cx*PX + Cy*PY + C0.
// ---------------------------------------------------------------------------
__global__ void face_prep_kernel(const float* __restrict__ verts,
                                 const int* __restrict__ faces,
                                 float4* __restrict__ wsA, int F, int FPAD) {
  int f = blockIdx.x * blockDim.x + threadIdx.x;
  if (f >= FPAD) return;
  if (f >= F) {  // padding faces: coefs 0, invArea 0 -> never inside
    float4 zf = make_float4(0.f, 0.f, 0.f, 0.f);
    wsA[0 * FPAD + f] = zf;
    wsA[1 * FPAD + f] = zf;
    wsA[2 * FPAD + f] = zf;
    wsA[3 * FPAD + f] = zf;
    return;
  }
  int i0 = faces[3 * f + 0], i1 = faces[3 * f + 1], i2 = faces[3 * f + 2];
  float x0 = verts[3 * i0], y0 = verts[3 * i0 + 1], z0 = verts[3 * i0 + 2];
  float x1 = verts[3 * i1], y1 = verts[3 * i1 + 1], z1 = verts[3 * i1 + 2];
  float x2 = verts[3 * i2], y2 = verts[3 * i2 + 1], z2 = verts[3 * i2 + 2];

  auto coef = [](float ax, float ay, float bx, float by) {
    float ex = bx - ax, ey = by - ay;
    // edge_fn = (bx-ax)*(PY-ay) - (by-ay)*(PX-ax) = -ey*PX + ex*PY + (ey*ax - ex*ay)
    return make_float4(-ey, ex, ey * ax - ex * ay, 0.0f);
  };
  wsA[0 * FPAD + f] = coef(x1, y1, x2, y2);  // w0
  wsA[1 * FPAD + f] = coef(x2, y2, x0, y0);  // w1
  wsA[2 * FPAD + f] = coef(x0, y0, x1, y1);  // w2

  float area = (x1 - x0) * (y2 - y0) - (y1 - y0) * (x2 - x0);
  float inv  = (fabsf(area) > EPSF) ? (1.0f / area) : 0.0f;  // 0 encodes "degenerate"
  wsA[3 * FPAD + f] = make_float4(z0, z1, z2, inv);
}

// ---------------------------------------------------------------------------
// Kernel 2: rasterize. One wave = 16-pixel row segment; 8 waves/block; 128 blocks.
// ---------------------------------------------------------------------------
__device__ __forceinline__ float edge_d2(float ax, float ay, float bx, float by,
                                         float px, float py) {
  float ex = bx - ax, ey = by - ay;
  float l2 = fmaxf(ex * ex + ey * ey, EPSF);
  float t = ((px - ax) * ex + (py - ay) * ey) / l2;
  t = fminf(fmaxf(t, 0.0f), 1.0f);
  float dx = px - (ax + t * ex);
  float dy = py - (ay + t * ey);
  return dx * dx + dy * dy;
}

__global__ __launch_bounds__(256)
void raster_main_kernel(const float* __restrict__ verts,
                        const int* __restrict__ faces,
                        int* __restrict__ outFace, float* __restrict__ outZ,
                        float* __restrict__ outBary, float* __restrict__ outDist,
                        const float4* __restrict__ wsA, int F, int FPAD) {
  __shared__ float4 sAll[4 * FPAD_MAX];  // 64512 B staged by the TDM

  const int lane = threadIdx.x & 31;
  const int wave = threadIdx.x >> 5;

  // ---- Tensor Data Mover: DMA all per-face data (4*FPAD float4) into LDS ----
  if (threadIdx.x < 32) {  // wave 0 issues one TDM op for the whole workgroup
    uint32_t ldsoff = (uint32_t)(uintptr_t)(void*)&sAll[0];  // low 32 bits = LDS offset
    uint64_t ga = (uint64_t)(uintptr_t)wsA;
    uint32_t nElem = (uint32_t)(4 * FPAD) * 4u;  // dwords (data_size = 4B)

    v4u g0;
    g0[0] = 1u;                                        // count=1, user descriptor
    g0[1] = ldsoff;                                    // lds_addr
    g0[2] = (uint32_t)ga;                              // global_addr[31:0]
    g0[3] = (uint32_t)((ga >> 32) & 0x01FFFFFFu) | (2u << 30);  // addr[56:32] | type=2

    v8i g1;
    g1[0] = (int)(2u << 16);                           // data_size = 4 bytes
    g1[1] = (int)((nElem & 0xFFFFu) << 16);            // tensor_dim0[15:0]
    g1[2] = (int)((nElem >> 16) & 0xFFFFu) | (1 << 16);// tensor_dim0[31:16] | tensor_dim1=1
    g1[3] = (int)((nElem & 0xFFFFu) << 16);            // tile_dim0 = nElem (fits 16 bits)
    g1[4] = 1;                                         // tile_dim1 = 1
    g1[5] = (int)nElem;                                // tensor_dim0_stride lo
    g1[6] = (int)((nElem & 0xFFFFu) << 16);            // stride hi=0 | tensor_dim1_stride lo
    g1[7] = 0;
    v4i g2 = {0, 0, 0, 0};
    v4i g3 = {0, 0, 0, 0};
    v8i g4 = {0, 0, 0, 0, 0, 0, 0, 0};
    __builtin_amdgcn_tensor_load_to_lds(g0, g1, g2, g3, g4, 0);
    __builtin_amdgcn_s_wait_tensorcnt(0);
  }
  __syncthreads();

  // ---- wave -> pixel tile mapping ----
  const int tile = blockIdx.x * 8 + wave;  // 0..1023
  const int ty = tile >> 3;                // image row h
  const int tx = tile & 7;                 // 16-pixel column tile
  const int lm = lane & 15;                // pixel within tile == face row within tile
  const bool lo = lane < 16;
  const int w_ix = tx * 16 + lm;
  const float px = 1.0f - 2.0f * ((float)w_ix + 0.5f) / (float)WW;
  const float py = 1.0f - 2.0f * ((float)ty + 0.5f) / (float)HH;

  // B operand (4x16, K=PX,PY,1,0 striped per pixel column)
  v2f bmat;
  bmat.x = lo ? px : 1.0f;
  bmat.y = lo ? py : 0.0f;

  // per-lane top-8 (this lane covers faces base+{0..7} (lo) or base+{8..15} (hi))
  float zlist[KK];
  int flist[KK];
#pragma unroll
  for (int k = 0; k < KK; ++k) { zlist[k] = BIGF; flist[k] = -1; }

  const v8f cz = {0.f, 0.f, 0.f, 0.f, 0.f, 0.f, 0.f, 0.f};
  const int nTiles = FPAD >> 4;

  for (int t = 0; t < nTiles; ++t) {
    const int base = t * 16;
    // A operands: per-edge face coefficients (16x4, M=face, K=Cx,Cy,C0,0)
    float4 c0 = sAll[0 * FPAD + base + lm];
    float4 c1 = sAll[1 * FPAD + base + lm];
    float4 c2 = sAll[2 * FPAD + base + lm];
    v2f a0; a0.x = lo ? c0.x : c0.z; a0.y = lo ? c0.y : c0.w;
    v2f a1; a1.x = lo ? c1.x : c1.z; a1.y = lo ? c1.y : c1.w;
    v2f a2; a2.x = lo ? c2.x : c2.z; a2.y = lo ? c2.y : c2.w;

    // D[face, pixel] = edge values for 16 faces x 16 pixels
    v8f d0 = __builtin_amdgcn_wmma_f32_16x16x4_f32(false, a0, false, bmat, (short)0, cz, false, false);
    v8f d1 = __builtin_amdgcn_wmma_f32_16x16x4_f32(false, a1, false, bmat, (short)0, cz, false, false);
    v8f d2 = __builtin_amdgcn_wmma_f32_16x16x4_f32(false, a2, false, bmat, (short)0, cz, false, false);

    const int fb = base + (lo ? 0 : 8);
#pragma unroll
    for (int r = 0; r < 8; ++r) {
      float4 fz = sAll[3 * FPAD + fb + r];  // (z0, z1, z2, invArea)
      float b0 = d0[r] * fz.w;
      float b1 = d1[r] * fz.w;
      float b2 = d2[r] * fz.w;
      bool inside = (b0 >= 0.f) && (b1 >= 0.f) && (b2 >= 0.f) && (fz.w != 0.f);
      float n0 = b0 * fz.y * fz.z;
      float n1 = fz.x * b1 * fz.z;
      float n2 = fz.x * fz.y * b2;
      float den = n0 + n1 + n2;
      float dens = (fabsf(den) < EPSF) ? EPSF : den;
      float zp = (n0 * fz.x + n1 * fz.y + n2 * fz.z) / dens;
      float zc = inside ? zp : BIGF;
      int fc = fb + r;
      // sorted insert (ascending z)
#pragma unroll
      for (int s = 0; s < KK; ++s) {
        bool sw = zc < zlist[s];
        float tz = zlist[s]; int tf = flist[s];
        zlist[s] = sw ? zc : zlist[s];
        flist[s] = sw ? fc : flist[s];
        zc = sw ? tz : zc;
        fc = sw ? tf : fc;
      }
    }
  }

  // ---- merge lane L with lane L^16 (both cover the same pixel) ----
  float pz[KK]; int pf[KK];
#pragma unroll
  for (int s = 0; s < KK; ++s) {
    pz[s] = __shfl(zlist[s], lane ^ 16, 32);
    pf[s] = __shfl(flist[s], lane ^ 16, 32);
  }
#pragma unroll
  for (int s = 0; s < KK; ++s) {
    float zc = pz[s]; int fc = pf[s];
#pragma unroll
    for (int q = 0; q < KK; ++q) {
      bool sw = zc < zlist[q];
      float tz = zlist[q]; int tf = flist[q];
      zlist[q] = sw ? zc : zlist[q];
      flist[q] = sw ? fc : flist[q];
      zc = sw ? tz : zc;
      fc = sw ? tf : fc;
    }
  }

  // ---- epilogue: lanes 0..15 own pixels; recompute bary/dists for the 8 winners ----
  if (lo) {
    const int pixBase = (ty * WW + w_ix) * KK;
#pragma unroll 1
    for (int k = 0; k < KK; ++k) {
      int f = flist[k];
      if (f >= 0 && f < F) {
        int i0 = faces[3 * f + 0], i1 = faces[3 * f + 1], i2 = faces[3 * f + 2];
        float x0 = verts[3 * i0], y0 = verts[3 * i0 + 1], z0 = verts[3 * i0 + 2];
        float x1 = verts[3 * i1], y1 = verts[3 * i1 + 1], z1 = verts[3 * i1 + 2];
        float x2 = verts[3 * i2], y2 = verts[3 * i2 + 1], z2 = verts[3 * i2 + 2];
        float w0 = (x2 - x1) * (py - y1) - (y2 - y1) * (px - x1);
        float w1 = (x0 - x2) * (py - y2) - (y0 - y2) * (px - x2);
        float w2 = (x1 - x0) * (py - y0) - (y1 - y0) * (px - x0);
        float area = (x1 - x0) * (y2 - y0) - (y1 - y0) * (x2 - x0);
        float asafe = (fabsf(area) < EPSF) ? EPSF : area;
        float b0 = w0 / asafe, b1 = w1 / asafe, b2 = w2 / asafe;
        bool inside = (b0 >= 0.f) && (b1 >= 0.f) && (b2 >= 0.f) && (fabsf(area) > EPSF);
        float n0 = b0 * z1 * z2, n1 = z0 * b1 * z2, n2 = z0 * z1 * b2;
        float den = n0 + n1 + n2;
        float dens = (fabsf(den) < EPSF) ? EPSF : den;
        float p0 = n0 / dens, p1 = n1 / dens, p2 = n2 / dens;
        float dd = fminf(edge_d2(x0, y0, x1, y1, px, py),
                   fminf(edge_d2(x1, y1, x2, y2, px, py),
                         edge_d2(x2, y2, x0, y0, px, py)));
        float sd = inside ? -dd : dd;
        outFace[pixBase + k] = f;
        outZ[pixBase + k] = zlist[k];
        outBary[(pixBase + k) * 3 + 0] = p0;
        outBary[(pixBase + k) * 3 + 1] = p1;
        outBary[(pixBase + k) * 3 + 2] = p2;
        outDist[pixBase + k] = sd;
      } else {
        outFace[pixBase + k] = -1;
        outZ[pixBase + k] = -1.0f;
        outBary[(pixBase + k) * 3 + 0] = -1.0f;
        outBary[(pixBase + k) * 3 + 1] = -1.0f;
        outBary[(pixBase + k) * 3 + 2] = -1.0f;
        outDist[pixBase + k] = -1.0f;
      }
    }
  }
}

// ---------------------------------------------------------------------------
extern "C" void kernel_launch(void* const* d_in, const int* in_sizes, int n_in,
                              void* d_out, int out_size, void* d_ws, size_t ws_size,
                              hipStream_t stream) {
  const float* verts = (const float*)d_in[0];  // (1, V, 3) f32
  const int* faces = (const int*)d_in[1];      // (1, F, 3) i32
  int F = in_sizes[1] / 3;
  int FPAD = ((F + 15) / 16) * 16;
  if (FPAD > FPAD_MAX) FPAD = FPAD_MAX;

  float4* wsA = (float4*)d_ws;  // 4 * FPAD * 16 bytes

  face_prep_kernel<<<(FPAD + 255) / 256, 256, 0, stream>>>(verts, faces, wsA, F, FPAD);

  const int N = HH * WW * KK;
  int* outFace = (int*)d_out;
  float* outZ = (float*)d_out + N;
  float* outBary = (float*)d_out + 2 * N;
  float* outDist = (float*)d_out + 5 * N;

  raster_main_kernel<<<(HH * WW / 16) / 8, 256, 0, stream>>>(
      verts, faces, outFace, outZ, outBary, outDist, wsA, F, FPAD);
}